// GraphTransformerWithPooling_50749333569685
// MI455X (gfx1250) — compile-verified
//
#include <hip/hip_runtime.h>

typedef __attribute__((ext_vector_type(2))) float v2f;
typedef __attribute__((ext_vector_type(8))) float v8f;

#define D_DIM 128
#define OUT_DIM 64
#define B_STRIDE 80  // float2 per K-pair row (64 + 16 pad): half-wave bank shift = 32

// Inner GEMM body: one wave computes MT x 16 rows x 64 cols.
// B fragments come from LDS in K-pair-interleaved layout: one ds_load_b64 each,
// reused for MT WMMAs.
template <bool RELU_IN, int MT>
__device__ __forceinline__ void gemm_body(
    const float* __restrict__ A, const float* __restrict__ bias,
    const float* __restrict__ sB, float* __restrict__ C, int rowBase,
    int colBase, int ncols, int lane) {
  const int half = lane >> 4;  // 0: holds K{0,1}; 1: holds K{2,3}
  const int lidx = lane & 15;

  const float* ap0 = A + (size_t)(rowBase + lidx) * D_DIM + 2 * half;
  const v2f* bbase = (const v2f*)sB + half * B_STRIDE + lidx;

  v8f acc[MT][4];
#pragma unroll
  for (int t = 0; t < 4; ++t) {
    const float bv = bias[colBase + t * 16 + lidx];  // N-only -> broadcast
#pragma unroll
    for (int m = 0; m < MT; ++m)
#pragma unroll
      for (int i = 0; i < 8; ++i) acc[m][t][i] = bv;
  }

#pragma unroll 8
  for (int k = 0; k < D_DIM; k += 4) {
    v2f a[MT];
#pragma unroll
    for (int m = 0; m < MT; ++m) {
      a[m] = *(const v2f*)(ap0 + (size_t)(m * 16) * D_DIM + k);  // 8B global
      if (RELU_IN) {
        a[m].x = fmaxf(a[m].x, 0.0f);
        a[m].y = fmaxf(a[m].y, 0.0f);
      }
    }
    const v2f* wk = bbase + (k >> 1) * B_STRIDE;
#pragma unroll
    for (int t = 0; t < 4; ++t) {
      const v2f b = wk[t * 16];  // one ds_load_b64: {W[k+2h][c], W[k+2h+1][c]}
#pragma unroll
      for (int m = 0; m < MT; ++m)
        acc[m][t] = __builtin_amdgcn_wmma_f32_16x16x4_f32(
            false, a[m], false, b, (short)0, acc[m][t], false, false);
    }
  }

#pragma unroll
  for (int m = 0; m < MT; ++m)
#pragma unroll
    for (int t = 0; t < 4; ++t) {
      const int col = colBase + t * 16 + lidx;
#pragma unroll
      for (int i = 0; i < 8; ++i) {
        const int row = rowBase + m * 16 + i + 8 * half;  // C: VGPR i = M i/i+8
        C[(size_t)row * ncols + col] = acc[m][t][i];
      }
    }
}

// C[M, ncols] = op(A[M,128]) @ W[128, ncols] + bias[ncols], op = identity/ReLU.
// blockDim = 256 (8 waves); block covers 256 rows x 64 cols.
template <bool RELU_IN>
__global__ __launch_bounds__(256) void gemm128_wmma_f32(
    const float* __restrict__ A, const float* __restrict__ W,
    const float* __restrict__ bias, float* __restrict__ C, int M, int ncols) {
  __shared__ float sB[64 * B_STRIDE * 2];  // 40 KB: K-pair interleaved W slice

  const int colBase = blockIdx.y * 64;

  // Stage W[:, colBase:colBase+64] as float2{W[2p][c], W[2p+1][c]}.
  for (int idx = threadIdx.x; idx < 64 * 16; idx += 256) {
    const int p = idx >> 4;          // K-pair row 0..63
    const int c = (idx & 15) << 2;   // col 0,4,...,60
    const float4 r0 = *(const float4*)(W + (size_t)(2 * p) * ncols + colBase + c);
    const float4 r1 =
        *(const float4*)(W + (size_t)(2 * p + 1) * ncols + colBase + c);
    float2* d = (float2*)&sB[(p * B_STRIDE + c) * 2];
    d[0] = make_float2(r0.x, r1.x);
    d[1] = make_float2(r0.y, r1.y);
    d[2] = make_float2(r0.z, r1.z);
    d[3] = make_float2(r0.w, r1.w);
  }
  __syncthreads();

  const int lane = threadIdx.x & 31;
  const int wave = threadIdx.x >> 5;
  const int rowBase = (blockIdx.x * 8 + wave) * 32;
  if (rowBase >= M) return;  // wave-uniform: EXEC all-ones for WMMA

  if (rowBase + 32 <= M)  // full 32-row tile
    gemm_body<RELU_IN, 2>(A, bias, sB, C, rowBase, colBase, ncols, lane);
  else  // 16-row tail (M % 32 == 16, M % 16 == 0)
    gemm_body<RELU_IN, 1>(A, bias, sB, C, rowBase, colBase, ncols, lane);
}

__global__ __launch_bounds__(256) void zero_f4(float4* __restrict__ p, int n4) {
  const int i = blockIdx.x * blockDim.x + threadIdx.x;
  if (i < n4) p[i] = make_float4(0.f, 0.f, 0.f, 0.f);
}

// out[dst[e]] += m[src[e]]  (row of 128 floats per edge).
// 32 threads per edge, float4 gather + 4 atomic adds; m/out are L2-resident.
__global__ __launch_bounds__(256) void scatter_add_edges(
    const float* __restrict__ m, const int* __restrict__ src,
    const int* __restrict__ dst, float* __restrict__ out, int E) {
  const int t = blockIdx.x * blockDim.x + threadIdx.x;
  const int e = t >> 5;
  if (e >= E) return;
  const int f = (t & 31) * 4;
  const int s = src[e];
  const int d = dst[e];
  const float4 v = *(const float4*)(m + (size_t)s * D_DIM + f);
  float* o = out + (size_t)d * D_DIM + f;
  atomicAdd(o + 0, v.x);
  atomicAdd(o + 1, v.y);
  atomicAdd(o + 2, v.z);
  atomicAdd(o + 3, v.w);
}

extern "C" void kernel_launch(void* const* d_in, const int* in_sizes, int n_in,
                              void* d_out, int out_size, void* d_ws,
                              size_t ws_size, hipStream_t stream) {
  const float* x = (const float*)d_in[0];
  const int* ei = (const int*)d_in[1];  // [2, E] int32
  const float* W1 = (const float*)d_in[2];
  const float* b1 = (const float*)d_in[3];
  const float* W2 = (const float*)d_in[4];
  const float* b2 = (const float*)d_in[5];
  const float* Wout = (const float*)d_in[6];
  const float* bout = (const float*)d_in[7];

  const int N = in_sizes[0] / D_DIM;
  const int E = in_sizes[1] / 2;
  const int* src = ei;
  const int* dst = ei + E;

  float* buf0 = (float*)d_ws;              // GEMM result m
  float* buf1 = buf0 + (size_t)N * D_DIM;  // scatter accumulator h

  const dim3 blk(256);
  const dim3 gridG((N + 255) / 256, D_DIM / 64);     // 128-col GEMMs
  const dim3 gridGO((N + 255) / 256, OUT_DIM / 64);  // 64-col output GEMM
  const int n4 = N * D_DIM / 4;
  const int zb = (n4 + 255) / 256;
  const int sb = (int)(((long long)E * 32 + 255) / 256);

  // ---- pool layer 1 (W1), iteration 1 ----
  gemm128_wmma_f32<false><<<gridG, blk, 0, stream>>>(x, W1, b1, buf0, N, D_DIM);
  zero_f4<<<zb, blk, 0, stream>>>((float4*)buf1, n4);
  scatter_add_edges<<<sb, blk, 0, stream>>>(buf0, src, dst, buf1, E);
  // ---- pool layer 1, iteration 2 ----
  gemm128_wmma_f32<false><<<gridG, blk, 0, stream>>>(buf1, W1, b1, buf0, N, D_DIM);
  zero_f4<<<zb, blk, 0, stream>>>((float4*)buf1, n4);
  scatter_add_edges<<<sb, blk, 0, stream>>>(buf0, src, dst, buf1, E);
  // ---- ReLU fused into next GEMM's A-load ----
  // ---- pool layer 2 (W2), iteration 1 ----
  gemm128_wmma_f32<true><<<gridG, blk, 0, stream>>>(buf1, W2, b2, buf0, N, D_DIM);
  zero_f4<<<zb, blk, 0, stream>>>((float4*)buf1, n4);
  scatter_add_edges<<<sb, blk, 0, stream>>>(buf0, src, dst, buf1, E);
  // ---- pool layer 2, iteration 2 ----
  gemm128_wmma_f32<false><<<gridG, blk, 0, stream>>>(buf1, W2, b2, buf0, N, D_DIM);
  zero_f4<<<zb, blk, 0, stream>>>((float4*)buf1, n4);
  scatter_add_edges<<<sb, blk, 0, stream>>>(buf0, src, dst, buf1, E);
  // ---- output projection with fused ReLU ----
  gemm128_wmma_f32<true><<<gridGO, blk, 0, stream>>>(buf1, Wout, bout,
                                                     (float*)d_out, N, OUT_DIM);
}